// GraphHallucinationRCNN_32392643346889
// MI455X (gfx1250) — compile-verified
//
#include <hip/hip_runtime.h>
#include <hip/hip_bf16.h>
#include <math.h>

// ---------------- problem constants ----------------
#define NOBJ 768
#define CF   1024
#define HC   256
#define KTOP 20
#define KK   21          // K + self
#define GH   256
#define H1H  4
#define XDIM 1028        // C + 4
#define NEG_SLOPE 0.2f
#define INV_IMG (1.0f / 800.0f)

typedef float v2f __attribute__((ext_vector_type(2)));
typedef float v8f __attribute__((ext_vector_type(8)));

// ---------------- workspace layout (floats) ----------------
// x / reg_in : [768,1028]
#define OFF_X      0
#define OFF_FA     (OFF_X + NOBJ * XDIM)              // [768,256]
#define OFF_FB     (OFF_FA + NOBJ * HC)               // [768,256]
#define OFF_REL    (OFF_FB + NOBJ * HC)               // [768,768]
#define OFF_XP1    (OFF_REL + NOBJ * NOBJ)            // [768,1024]
#define OFF_H1     (OFF_XP1 + NOBJ * CF)              // [768,1024]
#define OFF_ASRC1  (OFF_H1 + NOBJ * CF)               // [768,4]
#define OFF_ADST1  (OFF_ASRC1 + NOBJ * H1H)           // [768,4]
#define OFF_XP2    (OFF_ADST1 + NOBJ * H1H)           // [768,2]
#define OFF_A2S    (OFF_XP2 + NOBJ * 2)               // [768]
#define OFF_A2D    (OFF_A2S + NOBJ)                   // [768]
#define OFF_HID    (OFF_A2D + NOBJ)                   // [768,128]
#define OFF_NBR    (OFF_HID + NOBJ * 128)             // [768,21] int

__device__ __forceinline__ float waveReduceSum(float v) {
#pragma unroll
    for (int off = 16; off > 0; off >>= 1) v += __shfl_down(v, off, 32);
    return v;
}

__device__ __forceinline__ float leaky(float x) {
    return x >= 0.0f ? x : NEG_SLOPE * x;
}

// ---------------- build x = [feats | boxes/800] ----------------
__global__ __launch_bounds__(256) void build_x_kernel(const float* __restrict__ feats,
                                                      const float* __restrict__ boxes,
                                                      float* __restrict__ x) {
    int idx = blockIdx.x * 256 + threadIdx.x;
    if (idx >= NOBJ * XDIM) return;
    int i = idx / XDIM;
    int c = idx - i * XDIM;
    x[idx] = (c < CF) ? feats[i * CF + c] : boxes[i * 4 + (c - CF)] * INV_IMG;
}

// ---------------- generic fp32 WMMA GEMM: C[MxN] = A[MxK] * B[KxN] ----------------
// grid = (N/16, M/16), block = 32 (one wave per 16x16 tile), K % 4 == 0
__global__ __launch_bounds__(32) void wmma_gemm_f32(const float* __restrict__ A, int lda,
                                                    const float* __restrict__ B, int ldb,
                                                    float* __restrict__ C, int ldc,
                                                    int K,
                                                    const float* __restrict__ bias,
                                                    int do_relu) {
    const int lane = threadIdx.x;
    const int half = lane >> 4;       // 0: lanes 0-15, 1: lanes 16-31
    const int r    = lane & 15;
    const int n0   = blockIdx.x * 16;
    const int m0   = blockIdx.y * 16;

    v8f c = {0.f, 0.f, 0.f, 0.f, 0.f, 0.f, 0.f, 0.f};

    const float* arow = A + (m0 + r) * lda;
    for (int k0 = 0; k0 < K; k0 += 4) {
        int ka = k0 + 2 * half;               // this lane's K pair
        v2f a;
        a.x = arow[ka + 0];
        a.y = arow[ka + 1];
        v2f b;
        b.x = B[(ka + 0) * ldb + n0 + r];
        b.y = B[(ka + 1) * ldb + n0 + r];
        c = __builtin_amdgcn_wmma_f32_16x16x4_f32(false, a, false, b,
                                                  (short)0, c, false, false);
    }

#pragma unroll
    for (int v = 0; v < 8; ++v) {
        int row = m0 + v + 8 * half;
        int col = n0 + r;
        float val = c[v];
        if (bias) val += bias[col];
        if (do_relu) val = fmaxf(val, 0.0f);
        C[row * ldc + col] = val;
    }
}

// ---------------- pairwise relation scores ----------------
// rel[i,j] = sum_h relu(fa[i,h] + fb[j,h] + |bi-bj| @ Wg[:,h] + b1[h]) * W2[h] + b2
// one block (8 waves) per row i; wave w handles j = w, w+8, ...
__global__ __launch_bounds__(256) void rel_kernel(const float* __restrict__ fa,
                                                  const float* __restrict__ fb,
                                                  const float* __restrict__ boxes,
                                                  const float* __restrict__ Wg,   // [4,256]
                                                  const float* __restrict__ b1,   // [256]
                                                  const float* __restrict__ W2,   // [256]
                                                  const float* __restrict__ b2,   // [1]
                                                  float* __restrict__ rel) {
    __shared__ float fa_s[HC], w2_s[HC], b1_s[HC], wg_s[4 * HC];
    const int i = blockIdx.x;
    const int t = threadIdx.x;
    fa_s[t] = fa[i * HC + t];
    w2_s[t] = W2[t];
    b1_s[t] = b1[t];
#pragma unroll
    for (int d = 0; d < 4; ++d) wg_s[d * HC + t] = Wg[d * HC + t];
    __syncthreads();

    const float bx0 = boxes[i * 4 + 0], bx1 = boxes[i * 4 + 1];
    const float bx2 = boxes[i * 4 + 2], bx3 = boxes[i * 4 + 3];
    const float bias2 = b2[0];

    const int w = t >> 5;
    const int l = t & 31;
    for (int j = w; j < NOBJ; j += 8) {
        float d0 = fabsf(bx0 - boxes[j * 4 + 0]);
        float d1 = fabsf(bx1 - boxes[j * 4 + 1]);
        float d2 = fabsf(bx2 - boxes[j * 4 + 2]);
        float d3 = fabsf(bx3 - boxes[j * 4 + 3]);
        float acc = 0.0f;
#pragma unroll
        for (int s = 0; s < 8; ++s) {
            int h = l + 32 * s;
            float gw = d0 * wg_s[h] + d1 * wg_s[HC + h] + d2 * wg_s[2 * HC + h] + d3 * wg_s[3 * HC + h];
            float v = fa_s[h] + fb[j * HC + h] + gw + b1_s[h];
            v = fmaxf(v, 0.0f);
            acc += v * w2_s[h];
        }
        acc = waveReduceSum(acc);
        if (l == 0) {
            float rv = acc + bias2;
            if (j == i) rv = -1.0e9f;
            rel[i * NOBJ + j] = rv;
        }
    }
}

// ---------------- top-K=20 per row (iterative argmax, ties -> lower index) ----------------
__global__ __launch_bounds__(256) void topk_kernel(const float* __restrict__ rel,
                                                   int* __restrict__ nbr) {
    __shared__ float row_s[NOBJ];
    __shared__ float v_s[256];
    __shared__ int   i_s[256];
    const int i = blockIdx.x;
    const int t = threadIdx.x;
    for (int j = t; j < NOBJ; j += 256) row_s[j] = rel[i * NOBJ + j];
    __syncthreads();

    for (int k = 0; k < KTOP; ++k) {
        float bv = -3.4e38f;
        int   bi = 0x7fffffff;
        for (int j = t; j < NOBJ; j += 256) {
            float v = row_s[j];
            if (v > bv) { bv = v; bi = j; }
        }
        v_s[t] = bv; i_s[t] = bi;
        __syncthreads();
        for (int off = 128; off > 0; off >>= 1) {
            if (t < off) {
                float vo = v_s[t + off]; int io = i_s[t + off];
                if (vo > v_s[t] || (vo == v_s[t] && io < i_s[t])) { v_s[t] = vo; i_s[t] = io; }
            }
            __syncthreads();
        }
        if (t == 0) {
            nbr[i * KK + k] = i_s[0];
            row_s[i_s[0]] = -3.4e38f;
        }
        __syncthreads();
    }
    if (t == 0) nbr[i * KK + KTOP] = i;   // self loop
}

// ---------------- GAT1 attention scalars: asrc/adst [768,4] ----------------
__global__ __launch_bounds__(256) void att1_kernel(const float* __restrict__ xp1,
                                                   const float* __restrict__ g1as,  // [4,256]
                                                   const float* __restrict__ g1ad,  // [4,256]
                                                   float* __restrict__ asrc,
                                                   float* __restrict__ adst) {
    const int i = blockIdx.x;
    const int t = threadIdx.x;
    const int w = t >> 5;
    const int l = t & 31;
    const int h = w & 3;
    const int is_dst = w >> 2;
    const float* avec = is_dst ? g1ad : g1as;
    float acc = 0.0f;
#pragma unroll
    for (int s = 0; s < 8; ++s) {
        int f = l + 32 * s;
        acc += xp1[i * CF + h * GH + f] * avec[h * GH + f];
    }
    acc = waveReduceSum(acc);
    if (l == 0) {
        if (is_dst) adst[i * H1H + h] = acc;
        else        asrc[i * H1H + h] = acc;
    }
}

// ---------------- GAT1 softmax + aggregate -> h1 = relu(sum_k alpha*xp[nbr] + b) ----------------
__global__ __launch_bounds__(256) void gat1_agg_kernel(const float* __restrict__ xp1,
                                                       const float* __restrict__ asrc,
                                                       const float* __restrict__ adst,
                                                       const int* __restrict__ nbr,
                                                       const float* __restrict__ g1b,
                                                       float* __restrict__ h1) {
    __shared__ float alpha_s[KK * H1H];
    __shared__ int   nbrs[KK];
    const int i = blockIdx.x;
    const int t = threadIdx.x;
    if (t < KK) nbrs[t] = nbr[i * KK + t];
    __syncthreads();

    if (t < H1H) {
        const int h = t;
        const float ad = adst[i * H1H + h];
        float mx = -3.4e38f;
        for (int k = 0; k < KK; ++k) {
            float s = leaky(asrc[nbrs[k] * H1H + h] + ad);
            if (s > mx) mx = s;
        }
        float sum = 0.0f;
        for (int k = 0; k < KK; ++k) {
            float s = leaky(asrc[nbrs[k] * H1H + h] + ad);
            float e = __expf(s - mx);
            alpha_s[k * H1H + h] = e;
            sum += e;
        }
        float inv = 1.0f / sum;
        for (int k = 0; k < KK; ++k) alpha_s[k * H1H + h] *= inv;
    }
    __syncthreads();

#pragma unroll
    for (int s = 0; s < 4; ++s) {
        int c = t + 256 * s;
        int h = s;                      // c >> 8
        float acc = 0.0f;
        for (int k = 0; k < KK; ++k)
            acc += alpha_s[k * H1H + h] * xp1[nbrs[k] * CF + c];
        h1[i * CF + c] = fmaxf(acc + g1b[c], 0.0f);
    }
}

// ---------------- GAT2 projection: xp2 = h1 @ g2W  [768,2] ----------------
__global__ __launch_bounds__(64) void gat2_proj_kernel(const float* __restrict__ h1,
                                                       const float* __restrict__ g2W, // [1024,2]
                                                       float* __restrict__ xp2) {
    const int i = blockIdx.x;
    const int t = threadIdx.x;
    const int w = t >> 5;
    const int l = t & 31;
    float acc = 0.0f;
#pragma unroll 8
    for (int s = 0; s < 32; ++s) {
        int f = l + 32 * s;
        acc += h1[i * CF + f] * g2W[f * 2 + w];
    }
    acc = waveReduceSum(acc);
    if (l == 0) xp2[i * 2 + w] = acc;
}

// ---------------- GAT2 attention scalars ----------------
__global__ __launch_bounds__(256) void a2_kernel(const float* __restrict__ xp2,
                                                 const float* __restrict__ g2as, // [1,2]
                                                 const float* __restrict__ g2ad, // [1,2]
                                                 float* __restrict__ a2s,
                                                 float* __restrict__ a2d) {
    int i = blockIdx.x * 256 + threadIdx.x;
    if (i >= NOBJ) return;
    float v0 = xp2[i * 2 + 0], v1 = xp2[i * 2 + 1];
    a2s[i] = v0 * g2as[0] + v1 * g2as[1];
    a2d[i] = v0 * g2ad[0] + v1 * g2ad[1];
}

// ---------------- GAT2 aggregate -> out[:, 0:2] ----------------
__global__ __launch_bounds__(256) void gat2_agg_kernel(const float* __restrict__ xp2,
                                                       const float* __restrict__ a2s,
                                                       const float* __restrict__ a2d,
                                                       const int* __restrict__ nbr,
                                                       const float* __restrict__ g2b,
                                                       float* __restrict__ out) {
    int i = blockIdx.x * 256 + threadIdx.x;
    if (i >= NOBJ) return;
    const float ad = a2d[i];
    float mx = -3.4e38f;
    for (int k = 0; k < KK; ++k) {
        float s = leaky(a2s[nbr[i * KK + k]] + ad);
        if (s > mx) mx = s;
    }
    float sum = 0.0f;
    for (int k = 0; k < KK; ++k)
        sum += __expf(leaky(a2s[nbr[i * KK + k]] + ad) - mx);
    float inv = 1.0f / sum;
    float o0 = 0.0f, o1 = 0.0f;
    for (int k = 0; k < KK; ++k) {
        int j = nbr[i * KK + k];
        float w = __expf(leaky(a2s[j] + ad) - mx) * inv;
        o0 += w * xp2[j * 2 + 0];
        o1 += w * xp2[j * 2 + 1];
    }
    out[i * 6 + 0] = o0 + g2b[0];
    out[i * 6 + 1] = o1 + g2b[1];
}

// ---------------- copy h1 into reg_in (x buffer) first 1024 cols ----------------
__global__ __launch_bounds__(256) void copy_h1_kernel(const float* __restrict__ h1,
                                                      float* __restrict__ x) {
    int idx = blockIdx.x * 256 + threadIdx.x;   // 768*1024 elements
    int i = idx >> 10;
    int c = idx & 1023;
    x[i * XDIM + c] = h1[idx];
}

// ---------------- regression second layer + box decode -> out[:, 2:6] ----------------
__global__ __launch_bounds__(128) void reg2_decode_kernel(const float* __restrict__ hidden, // [768,128]
                                                          const float* __restrict__ rW2,    // [128,4]
                                                          const float* __restrict__ rb2,    // [4]
                                                          const float* __restrict__ boxes,
                                                          float* __restrict__ out) {
    __shared__ float sdelta[4];
    const int i = blockIdx.x;
    const int t = threadIdx.x;
    const int w = t >> 5;
    const int l = t & 31;
    float acc = 0.0f;
#pragma unroll
    for (int s = 0; s < 4; ++s) {
        int f = l + 32 * s;
        acc += hidden[i * 128 + f] * rW2[f * 4 + w];
    }
    acc = waveReduceSum(acc);
    if (l == 0) sdelta[w] = acc + rb2[w];
    __syncthreads();
    if (t == 0) {
        float x0 = boxes[i * 4 + 0], y0 = boxes[i * 4 + 1];
        float x1 = boxes[i * 4 + 2], y1 = boxes[i * 4 + 3];
        float pw = x1 - x0, ph = y1 - y0;
        float pcx = x0 + 0.5f * pw, pcy = y0 + 0.5f * ph;
        float dx = sdelta[0], dy = sdelta[1], dw = sdelta[2], dh = sdelta[3];
        float gcx = dx * pw + pcx, gcy = dy * ph + pcy;
        float gw = __expf(dw) * pw, gh = __expf(dh) * ph;
        out[i * 6 + 2] = gcx - 0.5f * gw;
        out[i * 6 + 3] = gcy - 0.5f * gh;
        out[i * 6 + 4] = gcx + 0.5f * gw;
        out[i * 6 + 5] = gcy + 0.5f * gh;
    }
}

extern "C" void kernel_launch(void* const* d_in, const int* in_sizes, int n_in,
                              void* d_out, int out_size, void* d_ws, size_t ws_size,
                              hipStream_t stream) {
    const float* feats = (const float*)d_in[0];
    const float* boxes = (const float*)d_in[1];
    const float* W1    = (const float*)d_in[2];
    const float* b1    = (const float*)d_in[3];
    const float* W2    = (const float*)d_in[4];
    const float* b2    = (const float*)d_in[5];
    const float* g1W   = (const float*)d_in[6];
    const float* g1as  = (const float*)d_in[7];
    const float* g1ad  = (const float*)d_in[8];
    const float* g1b   = (const float*)d_in[9];
    const float* g2W   = (const float*)d_in[10];
    const float* g2as  = (const float*)d_in[11];
    const float* g2ad  = (const float*)d_in[12];
    const float* g2b   = (const float*)d_in[13];
    const float* rW1   = (const float*)d_in[14];
    const float* rb1   = (const float*)d_in[15];
    const float* rW2   = (const float*)d_in[16];
    const float* rb2   = (const float*)d_in[17];

    float* ws    = (float*)d_ws;
    float* x     = ws + OFF_X;
    float* fa    = ws + OFF_FA;
    float* fb    = ws + OFF_FB;
    float* rel   = ws + OFF_REL;
    float* xp1   = ws + OFF_XP1;
    float* h1    = ws + OFF_H1;
    float* asrc1 = ws + OFF_ASRC1;
    float* adst1 = ws + OFF_ADST1;
    float* xp2   = ws + OFF_XP2;
    float* a2s   = ws + OFF_A2S;
    float* a2d   = ws + OFF_A2D;
    float* hid   = ws + OFF_HID;
    int*   nbr   = (int*)(ws + OFF_NBR);
    float* out   = (float*)d_out;

    // 1. x = [feats | geom]
    build_x_kernel<<<(NOBJ * XDIM + 255) / 256, 256, 0, stream>>>(feats, boxes, x);

    // 2/3. fa = feats @ W1[:1024], fb = feats @ W1[1024:2048]   (fp32 WMMA)
    wmma_gemm_f32<<<dim3(HC / 16, NOBJ / 16), 32, 0, stream>>>(feats, CF, W1, HC, fa, HC, CF, nullptr, 0);
    wmma_gemm_f32<<<dim3(HC / 16, NOBJ / 16), 32, 0, stream>>>(feats, CF, W1 + 1024 * HC, HC, fb, HC, CF, nullptr, 0);

    // 4. pairwise relation scores (diagonal masked)
    rel_kernel<<<NOBJ, 256, 0, stream>>>(fa, fb, boxes, W1 + 2048 * HC, b1, W2, b2, rel);

    // 5. top-K neighbors (+self)
    topk_kernel<<<NOBJ, 256, 0, stream>>>(rel, nbr);

    // 6. xp1 = x @ g1W   (768x1028x1024 fp32 WMMA)
    wmma_gemm_f32<<<dim3(CF / 16, NOBJ / 16), 32, 0, stream>>>(x, XDIM, g1W, CF, xp1, CF, XDIM, nullptr, 0);

    // 7. attention scalars
    att1_kernel<<<NOBJ, 256, 0, stream>>>(xp1, g1as, g1ad, asrc1, adst1);

    // 8. GAT1 aggregation -> h1
    gat1_agg_kernel<<<NOBJ, 256, 0, stream>>>(xp1, asrc1, adst1, nbr, g1b, h1);

    // 9. GAT2 projection
    gat2_proj_kernel<<<NOBJ, 64, 0, stream>>>(h1, g2W, xp2);

    // 10. GAT2 attention scalars
    a2_kernel<<<(NOBJ + 255) / 256, 256, 0, stream>>>(xp2, g2as, g2ad, a2s, a2d);

    // 11. GAT2 aggregation -> out[:, 0:2]
    gat2_agg_kernel<<<(NOBJ + 255) / 256, 256, 0, stream>>>(xp2, a2s, a2d, nbr, g2b, out);

    // 12. reg_in = [h1 | geom]   (reuse x buffer; geom tail already valid)
    copy_h1_kernel<<<(NOBJ * CF) / 256, 256, 0, stream>>>(h1, x);

    // 13. hidden = relu(reg_in @ rW1 + rb1)   (768x1028x128 fp32 WMMA, fused bias+relu)
    wmma_gemm_f32<<<dim3(128 / 16, NOBJ / 16), 32, 0, stream>>>(x, XDIM, rW1, 128, hid, 128, XDIM, rb1, 1);

    // 14. delta = hidden @ rW2 + rb2 ; box decode -> out[:, 2:6]
    reg2_decode_kernel<<<NOBJ, 128, 0, stream>>>(hid, rW2, rb2, boxes, out);
}